// NNUE_13666585936406
// MI455X (gfx1250) — compile-verified
//
#include <hip/hip_runtime.h>
#include <hip/hip_bf16.h>
#include <stddef.h>

// ---------------------------------------------------------------------------
// NNUE forward for MI455X (gfx1250, wave32).
// Strategy: exploit 0/1 sparsity of board features (exact f32 math),
// fold the factorizer into combined transposed weights (L2-resident),
// and run the MLP head on the matrix pipe with V_WMMA_F32_16X16X4_F32.
// ---------------------------------------------------------------------------

#define B_    1024
#define D_    49152
#define PD_   8192
#define BASE_ 160
#define PBASE_ 256

typedef __attribute__((ext_vector_type(2))) float v2f;
typedef __attribute__((ext_vector_type(8))) float v8f;

// ---------------------------------------------------------------------------
// Kernel 1: Wc[j, k] = W[k, j] + F[fmap[j], k]   (tiled transpose + factor add)
// grid: (C/32, D/32), block: (32, 8)
// ---------------------------------------------------------------------------
__global__ void combine_transpose(const float* __restrict__ W,   // [C][D]
                                  const float* __restrict__ F,   // [INTER][C]
                                  const int*   __restrict__ fmap,// [D]
                                  float* __restrict__ Wc,        // [D][C]
                                  int C, int D) {
    __shared__ float tile[32][33];
    const int k0 = blockIdx.x * 32;
    const int j0 = blockIdx.y * 32;
    const int tx = threadIdx.x;      // 0..31
    const int ty = threadIdx.y;      // 0..7
#pragma unroll
    for (int i = 0; i < 32; i += 8) {
        tile[ty + i][tx] = W[(size_t)(k0 + ty + i) * D + (j0 + tx)];
    }
    __syncthreads();
#pragma unroll
    for (int i = 0; i < 32; i += 8) {
        const int j = j0 + ty + i;
        const int f = fmap[j];
        Wc[(size_t)j * C + (k0 + tx)] =
            tile[tx][ty + i] + F[(size_t)f * C + (k0 + tx)];
    }
}

// ---------------------------------------------------------------------------
// Kernel 2: sparse feature transformer.
//   out[b, k] = bias[k] + sum_{j} x[b,j] * Wc[j, k]
// One block per batch row; blockDim.x == C (160 or 256, multiples of 32).
// Nonzeros compacted into an LDS list (ds_add_rtn_u32), then all C lanes
// stream coalesced Wc rows (L2 resident).
// ---------------------------------------------------------------------------
__global__ void ft_sparse(const float* __restrict__ x,    // [B][D]
                          const float* __restrict__ Wc,   // [D][C]
                          const float* __restrict__ bias, // [C]
                          float* __restrict__ out,        // [B][C]
                          int D) {
    __shared__ int   s_idx[1024];
    __shared__ float s_val[1024];
    __shared__ int   s_cnt;

    const int C   = blockDim.x;
    const int tid = threadIdx.x;
    const int b   = blockIdx.x;
    const float* xr = x + (size_t)b * D;

    float acc = bias[tid];
    const int span = C * 4;

    for (int base0 = 0; base0 < D; base0 += span) {
        if (tid == 0) s_cnt = 0;
        __syncthreads();
        const int j = base0 + tid * 4;
        if (j < D) {
            const float4 v = *reinterpret_cast<const float4*>(xr + j);
            if (v.x != 0.0f) { int p = atomicAdd(&s_cnt, 1); s_idx[p] = j + 0; s_val[p] = v.x; }
            if (v.y != 0.0f) { int p = atomicAdd(&s_cnt, 1); s_idx[p] = j + 1; s_val[p] = v.y; }
            if (v.z != 0.0f) { int p = atomicAdd(&s_cnt, 1); s_idx[p] = j + 2; s_val[p] = v.z; }
            if (v.w != 0.0f) { int p = atomicAdd(&s_cnt, 1); s_idx[p] = j + 3; s_val[p] = v.w; }
        }
        __syncthreads();
        const int cnt = s_cnt;
        for (int i = 0; i < cnt; ++i) {
            acc += s_val[i] * Wc[(size_t)s_idx[i] * C + tid];
        }
        __syncthreads();
    }
    out[(size_t)b * C + tid] = acc;
}

// ---------------------------------------------------------------------------
// Kernel 3: pov mixing + ReLU.
//   base   = relu(pov*[w_,b_]  + (1-pov)*[b_,w_])   [B][320]
//   p_base = relu(pov*[pw_,pb_]+ (1-pov)*[pb_,pw_]) [B][512]
// ---------------------------------------------------------------------------
__global__ void mix_kernel(const float* __restrict__ pov,
                           const float* __restrict__ w_,  const float* __restrict__ b_,
                           const float* __restrict__ pw_, const float* __restrict__ pb_,
                           float* __restrict__ base, float* __restrict__ pbase) {
    const int stride = gridDim.x * blockDim.x;
    const int t0 = blockIdx.x * blockDim.x + threadIdx.x;

    for (int idx = t0; idx < B_ * 2 * BASE_; idx += stride) {
        const int b = idx / (2 * BASE_);
        const int k = idx - b * (2 * BASE_);
        const float p = pov[b];
        const float c1 = (k < BASE_) ? w_[b * BASE_ + k] : b_[b * BASE_ + k - BASE_];
        const float c2 = (k < BASE_) ? b_[b * BASE_ + k] : w_[b * BASE_ + k - BASE_];
        const float v  = p * c1 + (1.0f - p) * c2;
        base[idx] = v > 0.0f ? v : 0.0f;
    }
    for (int idx = t0; idx < B_ * 2 * PBASE_; idx += stride) {
        const int b = idx / (2 * PBASE_);
        const int k = idx - b * (2 * PBASE_);
        const float p = pov[b];
        const float c1 = (k < PBASE_) ? pw_[b * PBASE_ + k] : pb_[b * PBASE_ + k - PBASE_];
        const float c2 = (k < PBASE_) ? pb_[b * PBASE_ + k] : pw_[b * PBASE_ + k - PBASE_];
        const float v  = p * c1 + (1.0f - p) * c2;
        pbase[idx] = v > 0.0f ? v : 0.0f;
    }
}

// ---------------------------------------------------------------------------
// WMMA helper: D += X[16,K] @ W.T  with W stored [16][K] row-major.
// Uses V_WMMA_F32_16X16X4_F32 per ISA 16x4 f32 A-layout:
//   lane L: m = L&15, hi = L>>4; VGPR r holds K = kstep + 2*hi + r.
// B (4x16) mirrors with n = L&15. C/D: VGPR r -> row (r + 8*hi), col = L&15.
// EXEC is all-ones at every call site (uniform control flow).
// ---------------------------------------------------------------------------
__device__ __forceinline__ v8f wmma_xk(const float* X, int ldx,
                                       const float* W, int K,
                                       v8f c, int lane) {
    const int m  = lane & 15;
    const int hi = lane >> 4;
    for (int k = 0; k < K; k += 4) {
        const v2f a  = *reinterpret_cast<const v2f*>(X + m * ldx + k + 2 * hi);
        const v2f bb = *reinterpret_cast<const v2f*>(W + m * K   + k + 2 * hi);
        c = __builtin_amdgcn_wmma_f32_16x16x4_f32(false, a, false, bb,
                                                  (short)0, c, false, false);
    }
    return c;
}

// ---------------------------------------------------------------------------
// Kernel 4: MLP head. 8 waves/block, one 16-row batch tile per wave.
// grid = 8 blocks (64 tiles of 16 rows).
// ---------------------------------------------------------------------------
__global__ __launch_bounds__(256) void head_kernel(
    const float* __restrict__ base,   // [B][320]
    const float* __restrict__ pbase,  // [B][512]
    const float* __restrict__ pfc0_W, const float* __restrict__ pfc0_b,
    const float* __restrict__ pfc1_W, const float* __restrict__ pfc1_b,
    const float* __restrict__ fc0_W,  const float* __restrict__ fc0_b,
    const float* __restrict__ fc1_W,  const float* __restrict__ fc1_b,
    const float* __restrict__ fc2_W,  const float* __restrict__ fc2_b,
    const float* __restrict__ fc3_W,  const float* __restrict__ fc3_b,
    float* __restrict__ out) {

    __shared__ __align__(16) float scat[8][16 * 48];   // [x0 | h1 | h2] per wave
    __shared__ __align__(16) float sp[8][16 * 16];     // p tile per wave

    const int wave = threadIdx.x >> 5;
    const int lane = threadIdx.x & 31;
    const int tile = blockIdx.x * 8 + wave;
    const int m0   = tile * 16;
    const int m    = lane & 15;
    const int hi   = lane >> 4;

    float* spw = sp[wave];
    float* sc  = scat[wave];
    const v8f z = {};

    // ---- p = relu(p_base @ pfc0_W.T + pfc0_b) : K = 512 ----
    v8f pf = wmma_xk(pbase + (size_t)m0 * 512, 512, pfc0_W, 512, z, lane);
    {
        const float bn = pfc0_b[m];
#pragma unroll
        for (int r = 0; r < 8; ++r) {
            float v = pf[r] + bn;
            spw[(r + 8 * hi) * 16 + m] = v > 0.0f ? v : 0.0f;
        }
    }
    __syncthreads();

    // ---- p2 = p @ pfc1_W.T (+ pfc1_b folded later) : K = 16 ----
    const v8f p2 = wmma_xk(spw, 16, pfc1_W, 16, z, lane);

    // ---- x0 = relu(base @ fc0_W.T + fc0_b + p2 + pfc1_b) : K = 320 ----
    v8f x0 = wmma_xk(base + (size_t)m0 * 320, 320, fc0_W, 320, z, lane);
    {
        const float bn = fc0_b[m] + pfc1_b[m];
#pragma unroll
        for (int r = 0; r < 8; ++r) {
            float v = x0[r] + p2[r] + bn;
            sc[(r + 8 * hi) * 48 + m] = v > 0.0f ? v : 0.0f;
        }
    }
    __syncthreads();

    // ---- h1 = relu(x0 @ fc1_W.T + fc1_b) : K = 16 ----
    v8f h1 = wmma_xk(sc, 48, fc1_W, 16, z, lane);
    {
        const float bn = fc1_b[m];
#pragma unroll
        for (int r = 0; r < 8; ++r) {
            float v = h1[r] + bn;
            sc[(r + 8 * hi) * 48 + 16 + m] = v > 0.0f ? v : 0.0f;
        }
    }
    __syncthreads();

    // ---- h2 = relu([x0,h1] @ fc2_W.T + fc2_b) : K = 32 ----
    v8f h2 = wmma_xk(sc, 48, fc2_W, 32, z, lane);
    {
        const float bn = fc2_b[m];
#pragma unroll
        for (int r = 0; r < 8; ++r) {
            float v = h2[r] + bn;
            sc[(r + 8 * hi) * 48 + 32 + m] = v > 0.0f ? v : 0.0f;
        }
    }
    __syncthreads();

    // ---- out = [x0,h1,h2] @ fc3_W.T + fc3_b : K = 48, N = 1 ----
    if (lane < 16) {
        float s = fc3_b[0];
#pragma unroll
        for (int k = 0; k < 48; ++k) s += sc[lane * 48 + k] * fc3_W[k];
        out[m0 + lane] = s;
    }
}

// ---------------------------------------------------------------------------
// Launch
// ---------------------------------------------------------------------------
extern "C" void kernel_launch(void* const* d_in, const int* in_sizes, int n_in,
                              void* d_out, int out_size, void* d_ws, size_t ws_size,
                              hipStream_t stream) {
    const float* pov     = (const float*)d_in[0];
    const float* white   = (const float*)d_in[1];
    const float* black   = (const float*)d_in[2];
    const float* p_white = (const float*)d_in[3];
    const float* p_black = (const float*)d_in[4];
    const int*   f_main  = (const int*)d_in[5];
    const int*   f_pawn  = (const int*)d_in[6];
    const float* aw_W = (const float*)d_in[7];
    const float* aw_b = (const float*)d_in[8];
    const float* aw_F = (const float*)d_in[9];
    const float* ab_W = (const float*)d_in[10];
    const float* ab_b = (const float*)d_in[11];
    const float* ab_F = (const float*)d_in[12];
    const float* pw_W = (const float*)d_in[13];
    const float* pw_b = (const float*)d_in[14];
    const float* pw_F = (const float*)d_in[15];
    const float* pb_W = (const float*)d_in[16];
    const float* pb_b = (const float*)d_in[17];
    const float* pb_F = (const float*)d_in[18];
    const float* fc0_W  = (const float*)d_in[19];
    const float* fc0_b  = (const float*)d_in[20];
    const float* pfc0_W = (const float*)d_in[21];
    const float* pfc0_b = (const float*)d_in[22];
    const float* pfc1_W = (const float*)d_in[23];
    const float* pfc1_b = (const float*)d_in[24];
    const float* fc1_W  = (const float*)d_in[25];
    const float* fc1_b  = (const float*)d_in[26];
    const float* fc2_W  = (const float*)d_in[27];
    const float* fc2_b  = (const float*)d_in[28];
    const float* fc3_W  = (const float*)d_in[29];
    const float* fc3_b  = (const float*)d_in[30];
    float* out = (float*)d_out;

    // workspace carve-up (floats). Total ~21.6M floats (~86.5 MB).
    float* ws    = (float*)d_ws;
    float* Wc_w  = ws;
    float* Wc_b  = Wc_w  + (size_t)D_ * BASE_;
    float* Wc_pw = Wc_b  + (size_t)D_ * BASE_;
    float* Wc_pb = Wc_pw + (size_t)PD_ * PBASE_;
    float* wbuf  = Wc_pb + (size_t)PD_ * PBASE_;   // [B][160]
    float* bbuf  = wbuf  + (size_t)B_ * BASE_;
    float* pwbuf = bbuf  + (size_t)B_ * BASE_;     // [B][256]
    float* pbbuf = pwbuf + (size_t)B_ * PBASE_;
    float* baseb = pbbuf + (size_t)B_ * PBASE_;    // [B][320]
    float* pbasb = baseb + (size_t)B_ * 2 * BASE_; // [B][512]

    const dim3 tb(32, 8);
    // 1) combined transposed weights (factorizer folded in)
    combine_transpose<<<dim3(BASE_ / 32, D_ / 32), tb, 0, stream>>>(aw_W, aw_F, f_main, Wc_w, BASE_, D_);
    combine_transpose<<<dim3(BASE_ / 32, D_ / 32), tb, 0, stream>>>(ab_W, ab_F, f_main, Wc_b, BASE_, D_);
    combine_transpose<<<dim3(PBASE_ / 32, PD_ / 32), tb, 0, stream>>>(pw_W, pw_F, f_pawn, Wc_pw, PBASE_, PD_);
    combine_transpose<<<dim3(PBASE_ / 32, PD_ / 32), tb, 0, stream>>>(pb_W, pb_F, f_pawn, Wc_pb, PBASE_, PD_);

    // 2) sparse feature transformers (bandwidth-bound on input read)
    ft_sparse<<<B_, BASE_,  0, stream>>>(white,   Wc_w,  aw_b, wbuf,  D_);
    ft_sparse<<<B_, BASE_,  0, stream>>>(black,   Wc_b,  ab_b, bbuf,  D_);
    ft_sparse<<<B_, PBASE_, 0, stream>>>(p_white, Wc_pw, pw_b, pwbuf, PD_);
    ft_sparse<<<B_, PBASE_, 0, stream>>>(p_black, Wc_pb, pb_b, pbbuf, PD_);

    // 3) pov mix + relu
    mix_kernel<<<512, 256, 0, stream>>>(pov, wbuf, bbuf, pwbuf, pbbuf, baseb, pbasb);

    // 4) MLP head on the matrix pipe (v_wmma_f32_16x16x4_f32)
    head_kernel<<<B_ / (16 * 8), 256, 0, stream>>>(
        baseb, pbasb,
        pfc0_W, pfc0_b, pfc1_W, pfc1_b,
        fc0_W, fc0_b, fc1_W, fc1_b, fc2_W, fc2_b, fc3_W, fc3_b,
        out);
}